// SelfFilterLoss_5884105195760
// MI455X (gfx1250) — compile-verified
//
#include <hip/hip_runtime.h>
#include <math.h>

// Problem shape (from reference)
#define BS_ROWS 16384
#define NCLS    8192
#define THREADS 256
#define WAVES   (THREADS / 32)
#define PER_THREAD (NCLS / THREADS)   // 32 floats per thread
#define NVEC       (PER_THREAD / 4)   // 8 x float4 per thread

#define LOG2E 1.4426950408889634f

typedef __attribute__((ext_vector_type(2))) float v2f;
typedef __attribute__((ext_vector_type(8))) float v8f;
typedef int v4i __attribute__((vector_size(4 * sizeof(int))));
typedef __attribute__((address_space(1))) v4i gv4i;   // global
typedef __attribute__((address_space(3))) v4i lv4i;   // LDS

#ifndef __has_builtin
#define __has_builtin(x) 0
#endif

#if __has_builtin(__builtin_amdgcn_global_load_async_to_lds_b128)
#define HAVE_ASYNC_LDS 1
#else
#define HAVE_ASYNC_LDS 0
#endif

#if __has_builtin(__builtin_amdgcn_wmma_f32_16x16x4_f32)
#define HAVE_WMMA_F32X4 1
#else
#define HAVE_WMMA_F32X4 0
#endif

// raw v_exp_f32 (base-2); inputs are bounded here so no libm clamping needed
__device__ __forceinline__ float fast_exp2(float x) {
#if __has_builtin(__builtin_amdgcn_exp2f)
  return __builtin_amdgcn_exp2f(x);
#else
  return __expf(x * 0.6931471805599453f);
#endif
}

__device__ __forceinline__ float fast_log(float x) {
#if __has_builtin(__builtin_amdgcn_logf)
  return __builtin_amdgcn_logf(x) * 0.6931471805599453f;  // log2 -> ln
#else
  return __logf(x);
#endif
}

// ---------- wave reductions (wave32) ----------

__device__ __forceinline__ float wave_sum(float v) {
#if HAVE_WMMA_F32X4
  // A(16x4): lane L<16 -> A[L][0]=v ; lane L>=16 -> A[L-16][2]=v ; col 1,3 = 0
  // B(4x16) = all ones  =>  D[m][n] = v_m + v_{m+16}
  v2f a; a[0] = v;    a[1] = 0.0f;
  v2f b; b[0] = 1.0f; b[1] = 1.0f;
  v8f c = {};
  v8f d = __builtin_amdgcn_wmma_f32_16x16x4_f32(false, a, false, b, (short)0, c,
                                                false, false);
  float p = ((d[0] + d[1]) + (d[2] + d[3])) + ((d[4] + d[5]) + (d[6] + d[7]));
  p += __shfl_xor(p, 16, 32);
  return p;   // full 32-lane sum, all lanes
#else
  #pragma unroll
  for (int off = 16; off > 0; off >>= 1) v += __shfl_xor(v, off, 32);
  return v;
#endif
}

__device__ __forceinline__ float wave_max(float v) {
  #pragma unroll
  for (int off = 16; off > 0; off >>= 1) v = fmaxf(v, __shfl_xor(v, off, 32));
  return v;
}

__device__ __forceinline__ float block_max(float v, float* red) {
  const int tid = threadIdx.x, lane = tid & 31, wid = tid >> 5;
  __syncthreads();
  v = wave_max(v);
  if (lane == 0) red[wid] = v;
  __syncthreads();
  float r = red[0];
  #pragma unroll
  for (int i = 1; i < WAVES; ++i) r = fmaxf(r, red[i]);  // uniform broadcast reads
  return r;
}

__device__ __forceinline__ void block_sum3(float& a, float& b, float& c,
                                           float (*red)[WAVES]) {
  const int tid = threadIdx.x, lane = tid & 31, wid = tid >> 5;
  __syncthreads();                       // protect red reuse
  a = wave_sum(a); b = wave_sum(b); c = wave_sum(c);
  if (lane == 0) { red[0][wid] = a; red[1][wid] = b; red[2][wid] = c; }
  __syncthreads();
  float ta = red[0][0], tb = red[1][0], tc = red[2][0];
  #pragma unroll
  for (int i = 1; i < WAVES; ++i) {
    ta += red[0][i]; tb += red[1][i]; tc += red[2][i];
  }
  a = ta; b = tb; c = tc;
}

__device__ __forceinline__ float block_sum(float v, float* red) {
  const int tid = threadIdx.x, lane = tid & 31, wid = tid >> 5;
  __syncthreads();
  v = wave_sum(v);
  if (lane == 0) red[wid] = v;
  __syncthreads();
  float r = red[0];
  #pragma unroll
  for (int i = 1; i < WAVES; ++i) r += red[i];
  return r;
}

// ---------- kernel 1: one block per row ----------

__global__ __launch_bounds__(THREADS) void self_filter_rows(
    const float* __restrict__ logits, const int* __restrict__ target,
    float* __restrict__ row_lpt, float* __restrict__ row_acc) {
  __shared__ float tile[NCLS];          // 32 KB row staging
  __shared__ float red3[3][WAVES];

  const int row = blockIdx.x;
  const int tid = threadIdx.x;
  const float* g = logits + (size_t)row * NCLS;

#if HAVE_ASYNC_LDS
  #pragma unroll
  for (int i = 0; i < NVEC; ++i) {
    const int idx = i * (THREADS * 4) + tid * 4;
    __builtin_amdgcn_global_load_async_to_lds_b128(
        (gv4i*)(unsigned long long)(g + idx),
        (lv4i*)(unsigned int)(unsigned long long)(tile + idx),
        0, 0);
  }
  #if __has_builtin(__builtin_amdgcn_s_wait_asynccnt)
  __builtin_amdgcn_s_wait_asynccnt(0);
  #else
  asm volatile("s_wait_asynccnt 0" ::: "memory");
  #endif
#else
  #pragma unroll
  for (int i = 0; i < NVEC; ++i) {
    const int idx = i * (THREADS * 4) + tid * 4;
    *(float4*)(tile + idx) = *(const float4*)(g + idx);
  }
#endif
  __syncthreads();

  // ---- pass 1: row max ----
  float m = -3.4e38f;
  #pragma unroll
  for (int i = 0; i < NVEC; ++i) {
    const int idx = i * (THREADS * 4) + tid * 4;
    float4 v = *(const float4*)(tile + idx);
    m = fmaxf(m, fmaxf(fmaxf(v.x, v.y), fmaxf(v.z, v.w)));
  }
  m = block_max(m, red3[0]);

  const float xt = tile[target[row]];            // uniform LDS broadcast
  const float K  = fast_exp2((m - xt) * LOG2E);  // exp(m - xt), per-row scalar

  // ---- pass 2 (fused): S0 = sum e, S1 = sum t*lat, S2 = sum lat ----
  // t = x - m ; e = exp(t) ; exp(x - xt) = e * K ; lat = max(0.2 - e*K, 0)
  float s0 = 0.0f, s1 = 0.0f, s2 = 0.0f;
  #pragma unroll
  for (int i = 0; i < NVEC; ++i) {
    const int idx = i * (THREADS * 4) + tid * 4;
    float4 v = *(const float4*)(tile + idx);
    #pragma unroll
    for (int k = 0; k < 4; ++k) {
      const float x   = (&v.x)[k];
      const float t   = x - m;
      const float e   = fast_exp2(t * LOG2E);
      const float lat = fmaxf(fmaf(-K, e, 0.2f), 0.0f);
      s0 += e;
      s2 += lat;
      s1 = fmaf(t, lat, s1);
    }
  }
  block_sum3(s0, s1, s2, red3);

  if (tid == 0) {
    const float L = fast_log(s0);
    row_lpt[row] = xt - m - L;        // log_prob at target
    row_acc[row] = s1 - L * s2;       // sum_j lp_j * latent_j
  }
}

// ---------- kernel 2: final deterministic reduction ----------

__global__ __launch_bounds__(THREADS) void self_filter_final(
    const float* __restrict__ row_lpt, const float* __restrict__ row_acc,
    float* __restrict__ out) {
  __shared__ float red[WAVES];
  const int tid = threadIdx.x;
  float s1 = 0.0f, s2 = 0.0f;
  for (int i = tid; i < BS_ROWS; i += THREADS) {
    s1 += row_lpt[i];
    s2 += row_acc[i];
  }
  s1 = block_sum(s1, red);
  s2 = block_sum(s2, red);
  if (tid == 0) {
    const float loss_ce  = -s1 / (float)BS_ROWS;
    const float one_loss = -s2 / (float)BS_ROWS;
    out[0] = loss_ce + 0.1f * one_loss / (float)(NCLS - 1);
  }
}

extern "C" void kernel_launch(void* const* d_in, const int* in_sizes, int n_in,
                              void* d_out, int out_size, void* d_ws, size_t ws_size,
                              hipStream_t stream) {
  (void)in_sizes; (void)n_in; (void)out_size; (void)ws_size;
  const float* logits = (const float*)d_in[0];
  const int*   target = (const int*)d_in[1];
  float* ws = (float*)d_ws;            // needs 2 * 16384 * 4 B = 128 KB
  float* row_lpt = ws;
  float* row_acc = ws + BS_ROWS;

  self_filter_rows<<<BS_ROWS, THREADS, 0, stream>>>(logits, target, row_lpt, row_acc);
  self_filter_final<<<1, THREADS, 0, stream>>>(row_lpt, row_acc, (float*)d_out);
}